// GNN_38594576122568
// MI455X (gfx1250) — compile-verified
//
#include <hip/hip_runtime.h>
#include <hip/hip_bf16.h>
#include <stdint.h>

#define HD 128
#define BN_EPS 1e-5f
#define LEAKY_SLOPE 0.01f

typedef float v2f __attribute__((ext_vector_type(2)));
typedef float v8f __attribute__((ext_vector_type(8)));

// ---------------------------------------------------------------- utilities
__global__ void zero_f32_kernel(float* __restrict__ p, long long n) {
    long long i = (long long)blockIdx.x * blockDim.x + threadIdx.x;
    if (i < n) p[i] = 0.f;
}

// x_op [N,7] -> [N,8] zero-padded
__global__ void pad_op_kernel(const float* __restrict__ x, float* __restrict__ o, int n) {
    int t = blockIdx.x * blockDim.x + threadIdx.x;
    if (t >= n * 8) return;
    int r = t >> 3, c = t & 7;
    o[t] = (c < 7) ? x[r * 7 + c] : 0.f;
}

// xm = x_m @ Wm + bm : [N,4]@[4,7] -> [N,8] zero-padded
__global__ void mtrans_kernel(const float* __restrict__ x, const float* __restrict__ Wm,
                              const float* __restrict__ bm, float* __restrict__ o, int n) {
    int t = blockIdx.x * blockDim.x + threadIdx.x;
    if (t >= n * 8) return;
    int r = t >> 3, c = t & 7;
    float acc = 0.f;
    if (c < 7) {
        acc = bm[c];
        #pragma unroll
        for (int k = 0; k < 4; ++k) acc += x[r * 4 + k] * Wm[k * 7 + c];
    }
    o[t] = acc;
}

// ---------------------------------------------------------------- scatter
// 128-wide gather + scatter-add: one wave per edge, float4 per lane
__global__ void scatter128_kernel(const float* __restrict__ feat, const int* __restrict__ ei,
                                  int E, float* __restrict__ agg) {
    long long t = (long long)blockIdx.x * blockDim.x + threadIdx.x;
    int e = (int)(t >> 5);
    int lane = (int)(t & 31);
    if (e >= E) return;
    int src = ei[e];
    int dst = ei[(long long)E + e];
    const float4 v = *(const float4*)(feat + (size_t)src * HD + lane * 4);
    float* o = agg + (size_t)dst * HD + lane * 4;
    atomicAdd(o + 0, v.x);
    atomicAdd(o + 1, v.y);
    atomicAdd(o + 2, v.z);
    atomicAdd(o + 3, v.w);
}

// 8-wide gather + scatter-add: 8 threads per edge
__global__ void scatter8_kernel(const float* __restrict__ feat, const int* __restrict__ ei,
                                int E, float* __restrict__ agg) {
    long long t = (long long)blockIdx.x * blockDim.x + threadIdx.x;
    int e = (int)(t >> 3);
    int f = (int)(t & 7);
    if (e >= E) return;
    int src = ei[e];
    int dst = ei[(long long)E + e];
    atomicAdd(&agg[(size_t)dst * 8 + f], feat[(size_t)src * 8 + f]);
}

// ---------------------------------------------------------------- BN stats
// column-wise sum / sumsq over h[N,128]; 128 threads/block, 256 rows/block
__global__ void colstats_kernel(const float* __restrict__ h, int N, float* __restrict__ sums) {
    int c = threadIdx.x;
    int r0 = blockIdx.x * 256;
    int rend = r0 + 256;
    if (rend > N) rend = N;
    float s = 0.f, s2 = 0.f;
    for (int r = r0; r < rend; ++r) {
        float v = h[(size_t)r * HD + c];
        s += v;
        s2 += v * v;
    }
    atomicAdd(&sums[c], s);
    atomicAdd(&sums[HD + c], s2);
}

// scale = g * rsqrt(var + eps); shift = be - mu*scale
__global__ void bnfinal_kernel(const float* __restrict__ sums, const float* __restrict__ g,
                               const float* __restrict__ be, float invN, float* __restrict__ ss) {
    int c = threadIdx.x;
    float mu = sums[c] * invN;
    float var = sums[HD + c] * invN - mu * mu;
    float sc = g[c] * rsqrtf(var + BN_EPS);
    ss[c] = sc;
    ss[HD + c] = be[c] - mu * sc;
}

// ---------------------------------------------------------------- GEMM (WMMA f32 16x16x4)
// Out[N,128] = op(A[N,Kpad]) @ W[Ka,128] + bias  (rows k in [Ka,Kpad) of W treated as 0)
// op(A) = A                          if BN_RELU == 0
//       = relu(A*ss[k] + ss[128+k])  if BN_RELU == 1   (per-K-column affine)
// ACC: Out += result ; LEAKY: leaky_relu epilogue.
// 256 threads = 8 waves; each wave computes a 16-row x 128-col strip.
// LDS W layout: row-pair interleaved -> element ((k>>1)*HD + n)*2 + (k&1),
// so each lane's (W[kk][n], W[kk+1][n]) is one aligned ds_load_b64.
template <int BN_RELU, int ACC, int LEAKY>
__launch_bounds__(256)
__global__ void gemm_wmma_kernel(const float* __restrict__ A, int lda,
                                 const float* __restrict__ W,
                                 const float* __restrict__ bias,
                                 const float* __restrict__ ss,
                                 float* __restrict__ Out,
                                 int N, int Ka, int Kpad) {
    extern __shared__ float lds[]; // Kpad*128 (paired W) + 256 (scale|shift)
    float* ldsW = lds;
    float* ldsS = lds + Kpad * HD;

    // stage weights into LDS, pair-interleaved, zero-pad rows >= Ka
    int total = Kpad * HD;
    for (int i = threadIdx.x; i < total; i += 256) {
        int kpair = i >> 8;            // i / (2*HD)
        int rem = i & (2 * HD - 1);
        int n = rem >> 1;
        int k = kpair * 2 + (rem & 1);
        ldsW[i] = (k < Ka) ? W[(size_t)k * HD + n] : 0.f;
    }
    if (BN_RELU) {
        if (threadIdx.x < 2 * HD) ldsS[threadIdx.x] = ss[threadIdx.x];
    }
    __syncthreads();

    int lane = threadIdx.x & 31;
    int wave = threadIdx.x >> 5;
    int r0 = (blockIdx.x * 8 + wave) * 16;

    // A row for this lane (ISA 16x4 f32 A layout: M = lane&15)
    int mrow = r0 + (lane & 15);
    if (mrow > N - 1) mrow = N - 1;          // clamp (results discarded at store)
    const float* arow = A + (size_t)mrow * lda;
    int nlo = lane & 15;
    int half = (lane < 16) ? 0 : 2;

    v8f c[8] = {};

    for (int k = 0; k < Kpad; k += 4) {
        // lanes 0-15 hold K = k,k+1 ; lanes 16-31 hold K = k+2,k+3
        int kk = k + half;
        v2f a = *(const v2f*)(arow + kk);
        if (BN_RELU) {
            v2f sc = *(const v2f*)&ldsS[kk];
            v2f sh = *(const v2f*)&ldsS[HD + kk];
            a.x = fmaxf(fmaf(a.x, sc.x, sh.x), 0.f);
            a.y = fmaxf(fmaf(a.y, sc.y, sh.y), 0.f);
        }
        // preload all 8 B pairs (issued as a ds clause, one wait), then 8 WMMAs
        const float* wrow = ldsW + (size_t)((kk >> 1) * HD) * 2;
        v2f b[8];
        #pragma unroll
        for (int t = 0; t < 8; ++t)
            b[t] = *(const v2f*)(wrow + (t * 16 + nlo) * 2);
        #pragma unroll
        for (int t = 0; t < 8; ++t)
            c[t] = __builtin_amdgcn_wmma_f32_16x16x4_f32(
                false, a, false, b[t], (short)0, c[t], false, false);
    }

    // C/D layout: VGPR i -> row r0 + i (+8 for lanes 16-31), col = lane&15 within tile
    int rbase = r0 + ((lane < 16) ? 0 : 8);
    if (r0 + 16 <= N) {
        // fast path: whole 16-row strip in bounds, no per-element guards
        #pragma unroll
        for (int t = 0; t < 8; ++t) {
            int col = t * 16 + nlo;
            float bv = bias[col];
            #pragma unroll
            for (int i = 0; i < 8; ++i) {
                size_t idx = (size_t)(rbase + i) * HD + col;
                float v = c[t][i] + bv;
                if (ACC) v += Out[idx];
                if (LEAKY) v = (v > 0.f) ? v : LEAKY_SLOPE * v;
                Out[idx] = v;
            }
        }
    } else {
        #pragma unroll
        for (int t = 0; t < 8; ++t) {
            int col = t * 16 + nlo;
            float bv = bias[col];
            #pragma unroll
            for (int i = 0; i < 8; ++i) {
                int row = rbase + i;
                if (row < N) {
                    size_t idx = (size_t)row * HD + col;
                    float v = c[t][i] + bv;
                    if (ACC) v += Out[idx];
                    if (LEAKY) v = (v > 0.f) ? v : LEAKY_SLOPE * v;
                    Out[idx] = v;
                }
            }
        }
    }
}

// ---------------------------------------------------------------- host
static inline int cdiv(long long a, long long b) { return (int)((a + b - 1) / b); }

extern "C" void kernel_launch(void* const* d_in, const int* in_sizes, int n_in,
                              void* d_out, int out_size, void* d_ws, size_t ws_size,
                              hipStream_t stream) {
    const float* x_op = (const float*)d_in[0];
    const float* x_m  = (const float*)d_in[1];
    const float* Wm   = (const float*)d_in[2];
    const float* bm   = (const float*)d_in[3];
    const float* W1_0 = (const float*)d_in[4];
    const float* b1_0 = (const float*)d_in[5];
    const float* g_0  = (const float*)d_in[6];
    const float* be_0 = (const float*)d_in[7];
    const float* W2_0 = (const float*)d_in[8];
    const float* b2_0 = (const float*)d_in[9];
    const float* W1_r = (const float*)d_in[10];
    const float* b1_r = (const float*)d_in[11];
    const float* g_r  = (const float*)d_in[12];
    const float* be_r = (const float*)d_in[13];
    const float* W2_r = (const float*)d_in[14];
    const float* b2_r = (const float*)d_in[15];
    const float* Wop  = (const float*)d_in[16];
    const float* bop  = (const float*)d_in[17];
    const float* Wm2  = (const float*)d_in[18];
    const float* bm2  = (const float*)d_in[19];
    const int* ei_oo  = (const int*)d_in[20];
    const int* ei_om  = (const int*)d_in[21];
    const int* ei_mo  = (const int*)d_in[22];
    const int* ei_mm  = (const int*)d_in[23];

    const int NOP = in_sizes[0] / 7;
    const int NM  = in_sizes[1] / 4;
    const int Eoo = in_sizes[20] / 2;
    const int Eom = in_sizes[21] / 2;
    const int Emo = in_sizes[22] / 2;
    const int Emm = in_sizes[23] / 2;

    // workspace carve-up
    float* w = (float*)d_ws;
    size_t off = 0;
    auto alloc = [&](size_t n) { float* p = w + off; off += n; return p; };
    float* xop_c  = alloc((size_t)NOP * HD);  // current / next op features (in-place)
    float* xm_c   = alloc((size_t)NM * HD);
    float* a_oo   = alloc((size_t)NOP * HD);  // agg buffers, also hold H in-place
    float* a_mo   = alloc((size_t)NOP * HD);
    float* a_om   = alloc((size_t)NM * HD);
    float* a_mm   = alloc((size_t)NM * HD);
    float* xop8   = alloc((size_t)NOP * 8);
    float* xm8    = alloc((size_t)NM * 8);
    float* a_oo8  = alloc((size_t)NOP * 8);
    float* a_mo8  = alloc((size_t)NOP * 8);
    float* a_om8  = alloc((size_t)NM * 8);
    float* a_mm8  = alloc((size_t)NM * 8);
    float* sums   = alloc(2 * HD);
    float* ss     = alloc(2 * HD);
    (void)ws_size; (void)n_in; (void)out_size;

    auto zero = [&](float* p, long long n) {
        zero_f32_kernel<<<cdiv(n, 256), 256, 0, stream>>>(p, n);
    };

    // prologue: pad op features, m_trans_fc
    pad_op_kernel<<<cdiv((long long)NOP * 8, 256), 256, 0, stream>>>(x_op, xop8, NOP);
    mtrans_kernel<<<cdiv((long long)NM * 8, 256), 256, 0, stream>>>(x_m, Wm, bm, xm8, NM);

    for (int l = 0; l < 3; ++l) {
        const float *W1[4], *b1[4], *gg[4], *bb[4], *W2[4], *b2[4];
        for (int t = 0; t < 4; ++t) {
            if (l == 0) {
                W1[t] = W1_0 + (size_t)t * 7 * HD;  b1[t] = b1_0 + (size_t)t * HD;
                gg[t] = g_0 + (size_t)t * HD;       bb[t] = be_0 + (size_t)t * HD;
                W2[t] = W2_0 + (size_t)t * HD * HD; b2[t] = b2_0 + (size_t)t * HD;
            } else {
                size_t base = (size_t)(l - 1) * 4 + t;
                W1[t] = W1_r + base * HD * HD;      b1[t] = b1_r + base * HD;
                gg[t] = g_r + base * HD;            bb[t] = be_r + base * HD;
                W2[t] = W2_r + base * HD * HD;      b2[t] = b2_r + base * HD;
            }
        }
        const int Ka   = (l == 0) ? 7 : HD;
        const int Kpad = (l == 0) ? 8 : HD;

        // --- scatter-add per edge type
        if (l == 0) {
            zero(a_oo8, (long long)NOP * 8); zero(a_mo8, (long long)NOP * 8);
            zero(a_om8, (long long)NM * 8);  zero(a_mm8, (long long)NM * 8);
            scatter8_kernel<<<cdiv((long long)Eoo * 8, 256), 256, 0, stream>>>(xop8, ei_oo, Eoo, a_oo8);
            scatter8_kernel<<<cdiv((long long)Emo * 8, 256), 256, 0, stream>>>(xm8,  ei_mo, Emo, a_mo8);
            scatter8_kernel<<<cdiv((long long)Eom * 8, 256), 256, 0, stream>>>(xop8, ei_om, Eom, a_om8);
            scatter8_kernel<<<cdiv((long long)Emm * 8, 256), 256, 0, stream>>>(xm8,  ei_mm, Emm, a_mm8);
        } else {
            zero(a_oo, (long long)NOP * HD); zero(a_mo, (long long)NOP * HD);
            zero(a_om, (long long)NM * HD);  zero(a_mm, (long long)NM * HD);
            scatter128_kernel<<<cdiv((long long)Eoo * 32, 256), 256, 0, stream>>>(xop_c, ei_oo, Eoo, a_oo);
            scatter128_kernel<<<cdiv((long long)Emo * 32, 256), 256, 0, stream>>>(xm_c,  ei_mo, Emo, a_mo);
            scatter128_kernel<<<cdiv((long long)Eom * 32, 256), 256, 0, stream>>>(xop_c, ei_om, Eom, a_om);
            scatter128_kernel<<<cdiv((long long)Emm * 32, 256), 256, 0, stream>>>(xm_c,  ei_mm, Emm, a_mm);
        }

        // --- 4 MLPs: GEMM1 -> BN stats -> (BN+ReLU fused) GEMM2 (+sum +leaky)
        auto run_mlp = [&](const float* Ain, float* Hb, int N, int t, float* Xout, int acc) {
            size_t sh1 = ((size_t)Kpad * HD + 2 * HD) * sizeof(float);
            size_t sh2 = ((size_t)HD * HD + 2 * HD) * sizeof(float);
            gemm_wmma_kernel<0, 0, 0><<<cdiv(N, 128), 256, sh1, stream>>>(
                Ain, Kpad, W1[t], b1[t], nullptr, Hb, N, Ka, Kpad);
            zero(sums, 2 * HD);
            colstats_kernel<<<cdiv(N, 256), HD, 0, stream>>>(Hb, N, sums);
            bnfinal_kernel<<<1, HD, 0, stream>>>(sums, gg[t], bb[t], 1.f / (float)N, ss);
            if (acc) {
                gemm_wmma_kernel<1, 1, 1><<<cdiv(N, 128), 256, sh2, stream>>>(
                    Hb, HD, W2[t], b2[t], ss, Xout, N, HD, HD);
            } else {
                gemm_wmma_kernel<1, 0, 0><<<cdiv(N, 128), 256, sh2, stream>>>(
                    Hb, HD, W2[t], b2[t], ss, Xout, N, HD, HD);
            }
        };

        run_mlp((l == 0) ? a_oo8 : a_oo, a_oo, NOP, 0, xop_c, 0);
        run_mlp((l == 0) ? a_mo8 : a_mo, a_mo, NOP, 2, xop_c, 1);
        run_mlp((l == 0) ? a_om8 : a_om, a_om, NM, 1, xm_c, 0);
        run_mlp((l == 0) ? a_mm8 : a_mm, a_mm, NM, 3, xm_c, 1);
    }

    // final projections straight into d_out
    float* out_op = (float*)d_out;
    float* out_m  = out_op + (size_t)NOP * HD;
    size_t shf = ((size_t)HD * HD + 2 * HD) * sizeof(float);
    gemm_wmma_kernel<0, 0, 0><<<cdiv(NOP, 128), 256, shf, stream>>>(
        xop_c, HD, Wop, bop, nullptr, out_op, NOP, HD, HD);
    gemm_wmma_kernel<0, 0, 0><<<cdiv(NM, 128), 256, shf, stream>>>(
        xm_c, HD, Wm2, bm2, nullptr, out_m, NM, HD, HD);
}